// MyPostHead_41755672052097
// MI455X (gfx1250) — compile-verified
//
#include <hip/hip_runtime.h>
#include <hip/hip_bf16.h>

typedef __attribute__((ext_vector_type(2))) float v2f;
typedef __attribute__((ext_vector_type(8))) float v8f;

#define B_  2
#define A_  2
#define C_  10
#define H_  32
#define W_  32
#define HW_ (H_ * W_)
#define N_  2048            // H*W*A
#define NT_ (N_ / 16)       // 128 row-tiles
#define EPSV 1e-6f

#if defined(__gfx1250__) &&                                                  \
    __has_builtin(__builtin_amdgcn_global_load_async_to_lds_b128) &&         \
    __has_builtin(__builtin_amdgcn_s_wait_asynccnt)
#define HAVE_ASYNC 1
typedef int v4i_ __attribute__((vector_size(16)));
typedef __attribute__((address_space(1))) v4i_ gv4i;   // global int4
typedef __attribute__((address_space(3))) v4i_ lv4i;   // LDS int4
#else
#define HAVE_ASYNC 0
#endif

__device__ __forceinline__ float fast_ratio(float num, float den) {
#if defined(__gfx1250__) && __has_builtin(__builtin_amdgcn_rcpf)
    return num * __builtin_amdgcn_rcpf(den);   // v_rcp_f32 + v_mul_f32
#else
    return __fdividef(num, den);
#endif
}

// ---------------------------------------------------------------------------
// Kernel 1: decoded (B,N,7) -> BEV boxes (B,N,4) [x1,y1,x2,y2]
// ---------------------------------------------------------------------------
__global__ void bev_kernel(const float* __restrict__ decoded,
                           float* __restrict__ bev) {
    int idx = blockIdx.x * blockDim.x + threadIdx.x;
    if (idx >= B_ * N_) return;
    const float* d = decoded + (size_t)idx * 7;
    float x = d[0], y = d[1], dx = d[3], dy = d[4], yaw = d[6];
    const float PI = 3.14159265358979323846f;
    float normed = fabsf(yaw - floorf(yaw / PI + 0.5f) * PI);
    bool  swapwh = normed > (PI * 0.25f);
    float w = swapwh ? dy : dx;
    float h = swapwh ? dx : dy;
    float4 b;
    b.x = x - 0.5f * w;
    b.y = y - 0.5f * h;
    b.z = x + 0.5f * w;
    b.w = y + 0.5f * h;
    ((float4*)bev)[idx] = b;
}

// ---------------------------------------------------------------------------
// Kernel 2: gather pp_params (B,C*3,H,W) -> p0t/p1t in [b][c][n] layout
// ---------------------------------------------------------------------------
__global__ void gather_pp_kernel(const float* __restrict__ pp,
                                 float* __restrict__ p0t,
                                 float* __restrict__ p1t) {
    int idx = blockIdx.x * blockDim.x + threadIdx.x;
    if (idx >= B_ * C_ * N_) return;
    int n  = idx & (N_ - 1);
    int bc = idx >> 11;              // / N_
    int c  = bc % C_;
    int b  = bc / C_;
    int hw = n >> 1;                 // n / A_
    size_t base = (size_t)b * (C_ * 3 * HW_) + (size_t)(c * 3) * HW_ + hw;
    p0t[idx] = pp[base];
    p1t[idx] = pp[base + HW_];
}

// ---------------------------------------------------------------------------
// Kernel 3: per-(b,c) softmax of scores over the N boxes -> v[b][c][n]
// ---------------------------------------------------------------------------
__global__ __launch_bounds__(256) void colsoftmax_kernel(
    const float* __restrict__ scores, float* __restrict__ v) {
    int c = blockIdx.x % C_;
    int b = blockIdx.x / C_;
    int t = threadIdx.x;
    __shared__ float red[256];

    float vals[8];
    float mx = -3.4028235e38f;
#pragma unroll
    for (int i = 0; i < 8; ++i) {
        int n  = t + i * 256;
        int a  = n & (A_ - 1);
        int hw = n >> 1;
        float s = scores[(size_t)b * (A_ * C_ * HW_) +
                         (size_t)(a * C_ + c) * HW_ + hw];
        vals[i] = s;
        mx = fmaxf(mx, s);
    }
    red[t] = mx;
    __syncthreads();
    for (int s = 128; s > 0; s >>= 1) {
        if (t < s) red[t] = fmaxf(red[t], red[t + s]);
        __syncthreads();
    }
    mx = red[0];
    __syncthreads();

    float sum = 0.0f;
#pragma unroll
    for (int i = 0; i < 8; ++i) {
        vals[i] = __expf(vals[i] - mx);
        sum += vals[i];
    }
    red[t] = sum;
    __syncthreads();
    for (int s = 128; s > 0; s >>= 1) {
        if (t < s) red[t] += red[t + s];
        __syncthreads();
    }
    float inv = 1.0f / red[0];
#pragma unroll
    for (int i = 0; i < 8; ++i) {
        int n = t + i * 256;
        v[(size_t)blockIdx.x * N_ + n] = vals[i] * inv;
    }
}

// ---------------------------------------------------------------------------
// e = exp(iou(n,m) + bias), fast reciprocal for the IoU ratio
// ---------------------------------------------------------------------------
__device__ __forceinline__ float exp_score(const float4 bn, float area_n,
                                           const float4 bm, float bias) {
    float ltx = fmaxf(bn.x, bm.x), lty = fmaxf(bn.y, bm.y);
    float rbx = fminf(bn.z, bm.z), rby = fminf(bn.w, bm.w);
    float iw  = fmaxf(rbx - ltx, 0.0f), ih = fmaxf(rby - lty, 0.0f);
    float inter = iw * ih;
    float am    = (bm.z - bm.x) * (bm.w - bm.y);
    float uni   = fmaxf(area_n + am - inter, EPSV);
    return __expf(fast_ratio(inter, uni) + bias);
}

// ---------------------------------------------------------------------------
// Kernel 4: fused IoU + bias + softmax-weighted sum via V_WMMA_F32_16X16X4_F32
// All 8 waves of a block share one (b,c): stage bev/p1/v (48KB) in LDS once
// (async global->LDS), then each wave runs the m-loop on a 16-row tile with
// two independent WMMA accumulators.
// ---------------------------------------------------------------------------
__global__ __launch_bounds__(256) void attn_kernel(
    const float* __restrict__ bev, const float* __restrict__ v,
    const float* __restrict__ p0t, const float* __restrict__ p1t,
    float* __restrict__ out) {
    __shared__ __align__(16) float4 sbev[N_];   // 32 KB
    __shared__ __align__(16) float  sp1[N_];    //  8 KB
    __shared__ __align__(16) float  sv [N_];    //  8 KB
    __shared__ float snum[8][16];
    __shared__ float sden[8][16];

    const int t    = threadIdx.x;
    const int lane = t & 31;
    const int wv   = t >> 5;

    const int bcb  = blockIdx.x >> 4;                 // 16 blocks per (b,c)
    const int c    = bcb % C_;
    const int b    = bcb / C_;
    const int tile = ((blockIdx.x & 15) << 3) + wv;   // this wave's 16-row tile

    const float4* gbev = (const float4*)bev + (size_t)b * N_;
    const float*  gp1  = p1t + (size_t)(b * C_ + c) * N_;
    const float*  gv   = v   + (size_t)(b * C_ + c) * N_;

    // ---- stage the (b,c) working set into LDS ----
#if HAVE_ASYNC
#pragma unroll
    for (int i = 0; i < 8; ++i) {
        int idx = t + i * 256;
        __builtin_amdgcn_global_load_async_to_lds_b128(
            (gv4i*)(v4i_*)(gbev + idx), (lv4i*)(v4i_*)(sbev + idx), 0, 0);
    }
#pragma unroll
    for (int i = 0; i < 2; ++i) {
        int idx = t + i * 256;
        __builtin_amdgcn_global_load_async_to_lds_b128(
            (gv4i*)((v4i_*)gp1 + idx), (lv4i*)((v4i_*)sp1 + idx), 0, 0);
        __builtin_amdgcn_global_load_async_to_lds_b128(
            (gv4i*)((v4i_*)gv + idx), (lv4i*)((v4i_*)sv + idx), 0, 0);
    }
    __builtin_amdgcn_s_wait_asynccnt(0);
#else
#pragma unroll
    for (int i = 0; i < 8; ++i) {
        int idx = t + i * 256;
        sbev[idx] = gbev[idx];
    }
#pragma unroll
    for (int i = 0; i < 2; ++i) {
        int idx = t + i * 256;
        ((float4*)sp1)[idx] = ((const float4*)gp1)[idx];
        ((float4*)sv )[idx] = ((const float4*)gv )[idx];
    }
#endif
    __syncthreads();

    // ---- per-lane WMMA tile roles ----
    const int halfId = lane >> 4;        // A: K={0,1} vs K={2,3}
    const int M      = lane & 15;        // A-matrix row
    const int k0     = halfId << 1;
    const int col    = lane & 15;        // B-matrix column
    const float sel0 = (col == 0) ? 1.0f : 0.0f;   // numerator column
    const float sel1 = (col == 1) ? 1.0f : 0.0f;   // denominator column

    const int    n      = tile * 16 + M;
    const float4 bn     = sbev[n];
    const float  area_n = (bn.z - bn.x) * (bn.w - bn.y);
    const float  p0v    = p0t[(size_t)(b * C_ + c) * N_ + n];

    const float2* sp1v2 = (const float2*)sp1;
    const float2* svv2  = (const float2*)sv;

    v8f acc0 = {};
    v8f acc1 = {};
    for (int m0 = 0; m0 < N_; m0 += 8) {
        {   // chunk 0: m in [m0, m0+4)
            const int ma = m0 + k0;
            const int pi = (m0 >> 1) + halfId;
            const float4 ba = sbev[ma];
            const float4 bb = sbev[ma + 1];
            const float2 pp = sp1v2[pi];
            const float2 vp = svv2[pi];
            v2f amat;
            amat.x = exp_score(bn, area_n, ba, p0v * pp.x);
            amat.y = exp_score(bn, area_n, bb, p0v * pp.y);
            v2f bmat;
            bmat.x = fmaf(vp.x, sel0, sel1);
            bmat.y = fmaf(vp.y, sel0, sel1);
            acc0 = __builtin_amdgcn_wmma_f32_16x16x4_f32(
                false, amat, false, bmat, (short)0, acc0, false, false);
        }
        {   // chunk 1: m in [m0+4, m0+8)
            const int ma = m0 + 4 + k0;
            const int pi = (m0 >> 1) + 2 + halfId;
            const float4 ba = sbev[ma];
            const float4 bb = sbev[ma + 1];
            const float2 pp = sp1v2[pi];
            const float2 vp = svv2[pi];
            v2f amat;
            amat.x = exp_score(bn, area_n, ba, p0v * pp.x);
            amat.y = exp_score(bn, area_n, bb, p0v * pp.y);
            v2f bmat;
            bmat.x = fmaf(vp.x, sel0, sel1);
            bmat.y = fmaf(vp.y, sel0, sel1);
            acc1 = __builtin_amdgcn_wmma_f32_16x16x4_f32(
                false, amat, false, bmat, (short)0, acc1, false, false);
        }
    }
    v8f acc = acc0 + acc1;

    // D layout: VGPR i -> (row i, col=lane) lanes 0-15 ; (row i+8) lanes 16-31.
    // numerator = column 0 (lanes 0,16), denominator = column 1 (lanes 1,17).
    if (lane == 0) {
#pragma unroll
        for (int i = 0; i < 8; ++i) snum[wv][i] = acc[i];
    } else if (lane == 16) {
#pragma unroll
        for (int i = 0; i < 8; ++i) snum[wv][i + 8] = acc[i];
    } else if (lane == 1) {
#pragma unroll
        for (int i = 0; i < 8; ++i) sden[wv][i] = acc[i];
    } else if (lane == 17) {
#pragma unroll
        for (int i = 0; i < 8; ++i) sden[wv][i + 8] = acc[i];
    }
    __syncthreads();   // uniform trip counts across all 8 waves

    if (lane < 16) {
        float o  = snum[wv][M] / sden[wv][M];
        int   nn = tile * 16 + M;
        int   a  = nn & 1;
        int   hw = nn >> 1;
        out[(size_t)((b * A_ + a) * C_ + c) * HW_ + hw] = o;
    }
}

// ---------------------------------------------------------------------------
extern "C" void kernel_launch(void* const* d_in, const int* in_sizes, int n_in,
                              void* d_out, int out_size, void* d_ws, size_t ws_size,
                              hipStream_t stream) {
    const float* scores  = (const float*)d_in[0];  // (2, 20, 32, 32)
    const float* bbox    = (const float*)d_in[1];  // (2, 14, 32, 32)
    const float* pp      = (const float*)d_in[2];  // (2, 30, 32, 32)
    const float* decoded = (const float*)d_in[3];  // (2, 2048, 7)
    float* out = (float*)d_out;

    float* ws  = (float*)d_ws;
    float* bev = ws;                              // B*N*4   = 16384 f32
    float* v   = ws + 16384;                      // B*C*N   = 40960 f32
    float* p0t = ws + 16384 + 40960;              // 40960 f32
    float* p1t = ws + 16384 + 2 * 40960;          // 40960 f32

    hipLaunchKernelGGL(bev_kernel, dim3((B_ * N_ + 255) / 256), dim3(256), 0,
                       stream, decoded, bev);
    hipLaunchKernelGGL(gather_pp_kernel, dim3((B_ * C_ * N_ + 255) / 256),
                       dim3(256), 0, stream, pp, p0t, p1t);
    hipLaunchKernelGGL(colsoftmax_kernel, dim3(B_ * C_), dim3(256), 0, stream,
                       scores, v);
    hipLaunchKernelGGL(attn_kernel, dim3(B_ * C_ * NT_ / 8), dim3(256), 0,
                       stream, bev, v, p0t, p1t, out);

    // pass-through outputs: bbox_preds then pp_params
    (void)hipMemcpyAsync(out + B_ * A_ * C_ * HW_, bbox,
                         sizeof(float) * (size_t)(B_ * A_ * 7 * HW_),
                         hipMemcpyDeviceToDevice, stream);
    (void)hipMemcpyAsync(out + B_ * A_ * C_ * HW_ + B_ * A_ * 7 * HW_, pp,
                         sizeof(float) * (size_t)(B_ * C_ * 3 * HW_),
                         hipMemcpyDeviceToDevice, stream);
}